// edGNNLayer_46858093199900
// MI455X (gfx1250) — compile-verified
//
#include <hip/hip_runtime.h>
#include <hip/hip_bf16.h>

typedef float v2f __attribute__((ext_vector_type(2)));
typedef float v8f __attribute__((ext_vector_type(8)));

#define N_NODES   50000
#define N_EDGES   800000
#define NODE_DIM  128
#define EDGE_DIM  128
#define OUT_FEATS 128
#define AGG_DIM   256            // agg_src(128) ++ agg_edge(128)
#define K_DIM     384            // NODE_DIM + AGG_DIM
#define LDS_STRIDE 388           // padded row stride (floats): 388 % 64 == 4 -> no bank conflicts
#define MTILE     32             // rows per GEMM block (2 WMMA tiles / wave)

// ---------------------------------------------------------------------------
// Kernel 1: zero the per-node edge counters.
// ---------------------------------------------------------------------------
__global__ void edgnn_zero_cnt(int* __restrict__ cnt, int n) {
    int i = blockIdx.x * blockDim.x + threadIdx.x;
    if (i < n) cnt[i] = 0;
}

// ---------------------------------------------------------------------------
// Kernel 2: histogram of destination nodes (800K int atomics on 200KB -> L2).
// ---------------------------------------------------------------------------
__global__ void edgnn_hist(const int* __restrict__ dst, int* __restrict__ cnt) {
    int e = blockIdx.x * blockDim.x + threadIdx.x;
    if (e < N_EDGES) atomicAdd(&cnt[dst[e]], 1);
}

// ---------------------------------------------------------------------------
// Kernel 3: single-block exclusive prefix scan over 50000 counts.
// Writes offs[0..N_NODES] and a cursor copy for the fill pass.
// ---------------------------------------------------------------------------
__global__ void __launch_bounds__(1024)
edgnn_scan(const int* __restrict__ cnt, int* __restrict__ offs,
           int* __restrict__ cursor) {
    __shared__ int buf[1024];
    __shared__ int base_s;
    const int tid = threadIdx.x;
    if (tid == 0) base_s = 0;
    __syncthreads();

    for (int chunk = 0; chunk < N_NODES; chunk += 1024) {
        int i = chunk + tid;
        int v = (i < N_NODES) ? cnt[i] : 0;
        buf[tid] = v;
        __syncthreads();
        // Hillis-Steele inclusive scan over the 1024-chunk
        #pragma unroll
        for (int ofs = 1; ofs < 1024; ofs <<= 1) {
            int t = (tid >= ofs) ? buf[tid - ofs] : 0;
            __syncthreads();
            buf[tid] += t;
            __syncthreads();
        }
        int incl = buf[tid];
        int base = base_s;
        if (i < N_NODES) {
            int excl = base + incl - v;
            offs[i]   = excl;
            cursor[i] = excl;
        }
        __syncthreads();                 // everyone has read base_s
        if (tid == 1023) base_s = base + incl;
        __syncthreads();
    }
    if (tid == 0) offs[N_NODES] = base_s;   // == N_EDGES
}

// ---------------------------------------------------------------------------
// Kernel 4: bucket fill -> edge ids grouped by destination node.
// ---------------------------------------------------------------------------
__global__ void edgnn_fill(const int* __restrict__ dst, int* __restrict__ cursor,
                           int* __restrict__ eidx) {
    int e = blockIdx.x * blockDim.x + threadIdx.x;
    if (e < N_EDGES) {
        int pos = atomicAdd(&cursor[dst[e]], 1);
        eidx[pos] = e;
    }
}

// ---------------------------------------------------------------------------
// Kernel 5: gather-aggregate. One wave32 per node: sum nf[src[e]] and ef[e]
// over incoming edges in registers (each lane owns 4+4 floats), one vector
// store. Zero float atomics. nf table (25.6MB) is L2-resident for gathers.
// ---------------------------------------------------------------------------
__global__ void __launch_bounds__(256)
edgnn_gather(const float* __restrict__ nf, const float* __restrict__ ef,
             const int* __restrict__ src, const int* __restrict__ offs,
             const int* __restrict__ eidx, float* __restrict__ agg) {
    int node = (blockIdx.x * blockDim.x + threadIdx.x) >> 5;
    int lane = threadIdx.x & 31;
    if (node >= N_NODES) return;

    int beg = offs[node];
    int end = offs[node + 1];

    float4 an = make_float4(0.f, 0.f, 0.f, 0.f);
    float4 ae = make_float4(0.f, 0.f, 0.f, 0.f);

    int j = beg;
    // two edges in flight to hide the eidx -> src -> row dependent-load chain
    for (; j + 1 < end; j += 2) {
        int e0 = eidx[j], e1 = eidx[j + 1];
        int s0 = src[e0], s1 = src[e1];
        float4 x0 = ((const float4*)(nf + (size_t)s0 * NODE_DIM))[lane];
        float4 y0 = ((const float4*)(ef + (size_t)e0 * EDGE_DIM))[lane];
        float4 x1 = ((const float4*)(nf + (size_t)s1 * NODE_DIM))[lane];
        float4 y1 = ((const float4*)(ef + (size_t)e1 * EDGE_DIM))[lane];
        an.x += x0.x; an.y += x0.y; an.z += x0.z; an.w += x0.w;
        ae.x += y0.x; ae.y += y0.y; ae.z += y0.z; ae.w += y0.w;
        an.x += x1.x; an.y += x1.y; an.z += x1.z; an.w += x1.w;
        ae.x += y1.x; ae.y += y1.y; ae.z += y1.z; ae.w += y1.w;
    }
    if (j < end) {
        int e0 = eidx[j];
        int s0 = src[e0];
        float4 x0 = ((const float4*)(nf + (size_t)s0 * NODE_DIM))[lane];
        float4 y0 = ((const float4*)(ef + (size_t)e0 * EDGE_DIM))[lane];
        an.x += x0.x; an.y += x0.y; an.z += x0.z; an.w += x0.w;
        ae.x += y0.x; ae.y += y0.y; ae.z += y0.z; ae.w += y0.w;
    }

    float* dstrow = agg + (size_t)node * AGG_DIM;
    ((float4*)dstrow)[lane] = an;
    ((float4*)(dstrow + NODE_DIM))[lane] = ae;
}

// ---------------------------------------------------------------------------
// Kernel 6: fused GEMM + bias + ReLU using V_WMMA_F32_16X16X4_F32.
// Block = 256 threads = 8 waves; block owns a 32-row M-tile (2 x 16-row WMMA
// tiles per wave, sharing each B fragment -> halves W L2 traffic). Wave w owns
// N-tile [16w,16w+16). A staged in LDS with padded stride. 2x96 WMMAs/wave.
//
// Fragment layouts (ISA 7.12.2, 32-bit operands):
//   A 16x4:  lane L -> row M=L&15, VGPR{0,1} = K = 2*(L>>4) + {0,1}
//   B 4x16:  lane L -> col N=L&15, VGPR{0,1} = K = 2*(L>>4) + {0,1}
//   C 16x16: lane L -> col N=L&15, VGPR r    = row M = 8*(L>>4) + r
// ---------------------------------------------------------------------------
__global__ void __launch_bounds__(256)
edgnn_gemm(const float* __restrict__ nf, const float* __restrict__ agg,
           const float* __restrict__ W, const float* __restrict__ bias,
           float* __restrict__ out) {
    __shared__ float Atile[MTILE * LDS_STRIDE];     // ~48.5 KB

    const int tid = threadIdx.x;
    const int m0  = blockIdx.x * MTILE;

    // Stage node-feature half: 32 rows x 32 float4 (clamp rows in last block)
    for (int i = tid; i < 1024; i += 256) {
        int r  = i >> 5;
        int c4 = i & 31;
        int gr = m0 + r; if (gr >= N_NODES) gr = N_NODES - 1;
        float4 v = ((const float4*)(nf + (size_t)gr * NODE_DIM))[c4];
        *(float4*)&Atile[r * LDS_STRIDE + c4 * 4] = v;
    }
    // Stage aggregate half: 32 rows x 64 float4
    for (int i = tid; i < 2048; i += 256) {
        int r  = i >> 6;
        int c4 = i & 63;
        int gr = m0 + r; if (gr >= N_NODES) gr = N_NODES - 1;
        float4 v = ((const float4*)(agg + (size_t)gr * AGG_DIM))[c4];
        *(float4*)&Atile[r * LDS_STRIDE + NODE_DIM + c4 * 4] = v;
    }
    __syncthreads();

    const int wv    = tid >> 5;          // N-tile 0..7
    const int lane  = tid & 31;
    const int nrow  = lane & 15;
    const int khalf = (lane >> 4) << 1;  // 0 or 2

    const float* arow0 = Atile + nrow * LDS_STRIDE;
    const float* arow1 = Atile + (16 + nrow) * LDS_STRIDE;
    const float* wrow  = W + (size_t)(wv * 16 + nrow) * K_DIM;   // B[k][n] = W[n][k]

    v8f c0 = {0.f, 0.f, 0.f, 0.f, 0.f, 0.f, 0.f, 0.f};
    v8f c1 = {0.f, 0.f, 0.f, 0.f, 0.f, 0.f, 0.f, 0.f};
    #pragma unroll 4
    for (int k = 0; k < K_DIM; k += 4) {
        v2f bf = *(const v2f*)(wrow  + k + khalf);
        v2f a0 = *(const v2f*)(arow0 + k + khalf);
        v2f a1 = *(const v2f*)(arow1 + k + khalf);
        c0 = __builtin_amdgcn_wmma_f32_16x16x4_f32(false, a0, false, bf,
                                                   (short)0, c0, false, false);
        c1 = __builtin_amdgcn_wmma_f32_16x16x4_f32(false, a1, false, bf,
                                                   (short)0, c1, false, false);
    }

    // Epilogue: bias + ReLU. Block-uniform fast path for full tiles (1562 of
    // 1563 blocks) -> one base address, 16 immediate-offset stores the
    // compiler can clause; guarded path only for the final partial tile.
    const int n     = wv * 16 + nrow;
    const float bv  = bias[n];
    const int mbase = m0 + ((lane >> 4) << 3);
    float* obase = out + (size_t)mbase * OUT_FEATS + n;

    if (m0 + MTILE <= N_NODES) {                 // uniform branch: full tile
        #pragma unroll
        for (int r = 0; r < 8; ++r) {
            float v0 = c0[r] + bv;
            float v1 = c1[r] + bv;
            obase[(size_t)r * OUT_FEATS]        = v0 > 0.f ? v0 : 0.f;
            obase[(size_t)(r + 16) * OUT_FEATS] = v1 > 0.f ? v1 : 0.f;
        }
    } else {                                     // last block: partial rows
        #pragma unroll
        for (int r = 0; r < 8; ++r) {
            if (mbase + r < N_NODES) {
                float v0 = c0[r] + bv;
                obase[(size_t)r * OUT_FEATS] = v0 > 0.f ? v0 : 0.f;
            }
            if (mbase + 16 + r < N_NODES) {
                float v1 = c1[r] + bv;
                obase[(size_t)(r + 16) * OUT_FEATS] = v1 > 0.f ? v1 : 0.f;
            }
        }
    }
}

// ---------------------------------------------------------------------------
// Host launcher
// ---------------------------------------------------------------------------
extern "C" void kernel_launch(void* const* d_in, const int* in_sizes, int n_in,
                              void* d_out, int out_size, void* d_ws, size_t ws_size,
                              hipStream_t stream) {
    const float* nf   = (const float*)d_in[0];   // [50000,128]
    const float* ef   = (const float*)d_in[1];   // [800000,128]
    const int*   src  = (const int*)d_in[2];     // [800000]
    const int*   dst  = (const int*)d_in[3];     // [800000]
    const float* W    = (const float*)d_in[4];   // [128,384]
    const float* bias = (const float*)d_in[5];   // [128]
    float*       out  = (float*)d_out;           // [50000,128]

    // Workspace layout (needs ~55.0 MB)
    char* ws = (char*)d_ws;
    float* agg    = (float*)ws;                                   // 51,200,000 B
    int*   cnt    = (int*)(ws + (size_t)N_NODES * AGG_DIM * 4);
    int*   offs   = cnt + N_NODES;                                // N_NODES+1 ints
    int*   cursor = offs + (N_NODES + 1);                         // N_NODES ints
    int*   eidx   = cursor + N_NODES;                             // N_EDGES ints

    // 1) zero counters
    edgnn_zero_cnt<<<(N_NODES + 255) / 256, 256, 0, stream>>>(cnt, N_NODES);
    // 2) histogram of dst
    edgnn_hist<<<(N_EDGES + 255) / 256, 256, 0, stream>>>(dst, cnt);
    // 3) exclusive scan -> offsets + cursors
    edgnn_scan<<<1, 1024, 0, stream>>>(cnt, offs, cursor);
    // 4) bucket fill -> edge ids grouped by dst
    edgnn_fill<<<(N_EDGES + 255) / 256, 256, 0, stream>>>(dst, cursor, eidx);
    // 5) per-node gather-aggregate (wave per node, no float atomics)
    edgnn_gather<<<N_NODES / 8, 256, 0, stream>>>(nf, ef, src, offs, eidx, agg);
    // 6) fused WMMA GEMM + bias + ReLU (32-row M-tiles)
    edgnn_gemm<<<(N_NODES + MTILE - 1) / MTILE, 256, 0, stream>>>(nf, agg, W, bias, out);
}